// FullyFluidLayer_28080496181270
// MI455X (gfx1250) — compile-verified
//
#include <hip/hip_runtime.h>
#include <hip/hip_bf16.h>

typedef __attribute__((ext_vector_type(2))) float v2f;
typedef __attribute__((ext_vector_type(8))) float v8f;

#define M_DIM 256
#define N_DIM 2048
#define K_DIM 8192
#define T_STEPS 50
#define DIFF 0.01f
#define VISC 0.001f

#define SCAN_T 256
#define EPT 8   // 256 threads * 8 = 2048 elements

// ---------------------------------------------------------------------------
// GEMM: P[256,2048] = X[256,8192] @ W[8192,2048], exact f32 via WMMA 16x16x4.
// Each wave computes a 16(M) x 64(N) tile with 4 accumulators.
// ---------------------------------------------------------------------------
__global__ __launch_bounds__(256) void fluid_gemm_wmma(const float* __restrict__ X,
                                                       const float* __restrict__ Wm,
                                                       float* __restrict__ P) {
    const int wave  = blockIdx.x * (blockDim.x >> 5) + (threadIdx.x >> 5);
    const int lane  = threadIdx.x & 31;
    const int half  = lane >> 4;   // 0: K pair {0,1}, 1: K pair {2,3}
    const int l16   = lane & 15;
    const int mTile = wave & 15;   // 16 tiles of M=16
    const int nGrp  = wave >> 4;   // 32 groups of N=64
    const int mBase = mTile * 16;
    const int nBase = nGrp * 64;

    v8f acc0 = {}, acc1 = {}, acc2 = {}, acc3 = {};

    // A (16x4 f32): lanes 0-15 hold M=l16 with VGPR0/1 = K {0,1};
    // lanes 16-31 hold K {2,3}. Contiguous pair -> one b64 load.
    const float* aRow = X + (size_t)(mBase + l16) * K_DIM + 2 * half;
    // B (4x16 f32): VGPR0 = row K (lanes 0-15) / K+2 (lanes 16-31), N = l16.
    const float* bCol = Wm + (nBase + l16);

    #pragma unroll 2
    for (int k = 0; k < K_DIM; k += 4) {
        v2f a = *(const v2f*)(aRow + k);
        const int r0 = (k + 2 * half) * N_DIM;
        const int r1 = r0 + N_DIM;
        v2f bv;
        bv.x = bCol[r0 +  0]; bv.y = bCol[r1 +  0];
        acc0 = __builtin_amdgcn_wmma_f32_16x16x4_f32(false, a, false, bv, (short)0, acc0, false, false);
        bv.x = bCol[r0 + 16]; bv.y = bCol[r1 + 16];
        acc1 = __builtin_amdgcn_wmma_f32_16x16x4_f32(false, a, false, bv, (short)0, acc1, false, false);
        bv.x = bCol[r0 + 32]; bv.y = bCol[r1 + 32];
        acc2 = __builtin_amdgcn_wmma_f32_16x16x4_f32(false, a, false, bv, (short)0, acc2, false, false);
        bv.x = bCol[r0 + 48]; bv.y = bCol[r1 + 48];
        acc3 = __builtin_amdgcn_wmma_f32_16x16x4_f32(false, a, false, bv, (short)0, acc3, false, false);
    }

    // C/D layout: VGPR r -> M = r (lanes 0-15) / r+8 (lanes 16-31), N = l16.
    const int m0 = mBase + half * 8;
    #pragma unroll
    for (int r = 0; r < 8; ++r) {
        float* orow = P + (size_t)(m0 + r) * N_DIM + nBase + l16;
        orow[ 0] = acc0[r];
        orow[16] = acc1[r];
        orow[32] = acc2[r];
        orow[48] = acc3[r];
    }
}

// ---------------------------------------------------------------------------
// Sequential fluid scan: 256 samples x 50 steps over a 2048-element state.
// One workgroup (8 waves). s/vx/vy in registers, 8 elements per thread.
// Only thread-boundary elements go through LDS; double-buffered edges give
// exactly one barrier per time step.
// ---------------------------------------------------------------------------
__global__ __launch_bounds__(SCAN_T) void fluid_scan(const float* __restrict__ P,
                                                     const float* __restrict__ vx0,
                                                     const float* __restrict__ vy0,
                                                     float* __restrict__ out) {
    __shared__ float edgeL[2][SCAN_T];   // each thread's element 0
    __shared__ float edgeR[2][SCAN_T];   // each thread's element EPT-1

    const int t    = threadIdx.x;
    const int base = t * EPT;

    float vx[EPT], vy[EPT];
    #pragma unroll
    for (int e = 0; e < EPT; ++e) { vx[e] = vx0[base + e]; vy[e] = vy0[base + e]; }

    for (int b = 0; b < M_DIM; ++b) {
        float s[EPT];
        const float4* p4 = (const float4*)(P + (size_t)b * N_DIM + base);
        float4 a0 = p4[0], a1 = p4[1];
        s[0] = a0.x; s[1] = a0.y; s[2] = a0.z; s[3] = a0.w;
        s[4] = a1.x; s[5] = a1.y; s[6] = a1.z; s[7] = a1.w;

        edgeL[0][t] = s[0];
        edgeR[0][t] = s[EPT - 1];
        __syncthreads();

        int cur = 0;
        for (int step = 0; step < T_STEPS; ++step) {
            // rp[i] = s[i-1], rm[i] = s[i+1] (circular)
            const float left  = edgeR[cur][(t + SCAN_T - 1) & (SCAN_T - 1)]; // s[base-1]
            const float right = edgeL[cur][(t + 1) & (SCAN_T - 1)];          // s[base+EPT]

            float ns[EPT];
            #pragma unroll
            for (int e = 0; e < EPT; ++e) {
                const float se  = s[e];
                const float sm1 = (e == 0)       ? left  : s[e - 1];
                const float sp1 = (e == EPT - 1) ? right : s[e + 1];
                const float dR = sp1 - se;      // rm - s
                const float dL = sm1 - se;      // rp - s
                float nvx = vx[e] - DIFF * dR;
                nvx = fminf(fmaxf(nvx, -1.0f), 1.0f);
                float nvy = vy[e] - DIFF * dL;
                nvy = fminf(fmaxf(nvy, -1.0f), 1.0f);
                vx[e] = nvx; vy[e] = nvy;
                const float lap = dL + dR;      // rp + rm - 2s
                float v = se + DIFF * lap - VISC * se - nvx * dR - nvy * dL;
                v = fminf(fmaxf(v, -10.0f), 10.0f);
                ns[e] = v;
            }

            const int nxt = cur ^ 1;
            edgeL[nxt][t] = ns[0];
            edgeR[nxt][t] = ns[EPT - 1];
            #pragma unroll
            for (int e = 0; e < EPT; ++e) s[e] = ns[e];
            cur = nxt;
            __syncthreads();   // single barrier per step (double-buffered edges)
        }

        float4 o0, o1;
        o0.x = s[0]; o0.y = s[1]; o0.z = s[2]; o0.w = s[3];
        o1.x = s[4]; o1.y = s[5]; o1.z = s[6]; o1.w = s[7];
        float4* q4 = (float4*)(out + (size_t)b * N_DIM + base);
        q4[0] = o0; q4[1] = o1;
    }
}

extern "C" void kernel_launch(void* const* d_in, const int* in_sizes, int n_in,
                              void* d_out, int out_size, void* d_ws, size_t ws_size,
                              hipStream_t stream) {
    const float* x   = (const float*)d_in[0];   // (256, 8192)
    const float* Wm  = (const float*)d_in[1];   // (8192, 2048)
    const float* vx0 = (const float*)d_in[2];   // (2048,)
    const float* vy0 = (const float*)d_in[3];   // (2048,)
    float* out = (float*)d_out;                 // (256, 2048)

    const size_t needP = (size_t)M_DIM * N_DIM * sizeof(float);
    float* P = (ws_size >= needP) ? (float*)d_ws : out;  // scan reads its own slots
                                                         // before overwriting: in-place safe

    // 512 wave-tiles (16 M-tiles x 32 N-groups), 8 waves per block -> 64 blocks
    fluid_gemm_wmma<<<64, 256, 0, stream>>>(x, Wm, P);
    // inherently sequential scan: one persistent workgroup
    fluid_scan<<<1, SCAN_T, 0, stream>>>(P, vx0, vy0, out);
}